// DeformableConv2d_49039936586387
// MI455X (gfx1250) — compile-verified
//
#include <hip/hip_runtime.h>

typedef __attribute__((ext_vector_type(2))) float v2f;
typedef __attribute__((ext_vector_type(8))) float v8f;

#define B_   32
#define C_   64
#define CO_  64
#define H_   64
#define W_   64
#define KK_  9
#define KDIM_ (C_ * KK_)   /* 576 */
#define TM_  16            /* pixels per tile */

__global__ __launch_bounds__(128)
void deform_conv2d_wmma_f32(const float* __restrict__ x,
                            const float* __restrict__ w_off,
                            const float* __restrict__ b_off,
                            const float* __restrict__ w_conv,
                            const float* __restrict__ b_conv,
                            float* __restrict__ out)
{
    __shared__ float sA[KDIM_ * TM_];       // sampled[k = c*9+kk][m], 36 KB
    __shared__ float sPatch[C_ * 3 * 18];   // x patch [c][row 0..2][col 0..17], 13.5 KB
    __shared__ float sOff[18 * TM_];        // offsets[oc][m], 1.1 KB

    const int tid  = threadIdx.x;
    const int tile = blockIdx.x;
    const int wt   = tile & 3;          // 16-wide strip in W
    const int ho   = (tile >> 2) & 63;
    const int b    = tile >> 8;
    const int w0   = wt * 16;

    // ---- Phase 0a: load zero-padded 3x18 x-patch for all 64 channels ----
    for (int i = tid; i < C_ * 3 * 18; i += 128) {
        int c   = i / 54;
        int rem = i - c * 54;
        int r   = rem / 18;
        int col = rem - r * 18;
        int y   = ho - 1 + r;
        int xx  = w0 - 1 + col;
        float v = 0.0f;
        if ((unsigned)y < (unsigned)H_ && (unsigned)xx < (unsigned)W_)
            v = x[(((size_t)b * C_ + c) * H_ + y) * W_ + xx];
        sPatch[i] = v;
    }
    __syncthreads();

    // ---- Phase 0b: offset conv (18 channels x 16 pixels, K-depth 576) ----
    for (int p = tid; p < 18 * TM_; p += 128) {
        int oc = p >> 4;
        int m  = p & 15;
        float s = b_off[oc];
        const float* wrow = w_off + (size_t)oc * KDIM_;
        for (int c = 0; c < C_; ++c) {
            const float* pr = sPatch + c * 54;
            const float* wc = wrow + c * KK_;
            #pragma unroll
            for (int ki = 0; ki < 3; ++ki)
                #pragma unroll
                for (int kj = 0; kj < 3; ++kj)
                    s += pr[ki * 18 + m + kj] * wc[ki * 3 + kj];
        }
        sOff[oc * 16 + m] = s;
    }
    __syncthreads();

    // ---- Phase A: bilinear sampling into sA (channel-invariant addresses) ----
    for (int p = tid; p < TM_ * KK_; p += 128) {
        int m  = p / KK_;
        int kk = p - m * KK_;
        int ki = kk / 3, kj = kk - ki * 3;
        float dy = sOff[(2 * kk)     * 16 + m];
        float dx = sOff[(2 * kk + 1) * 16 + m];
        float py = (float)(ho - 1 + ki) + dy;
        float px = (float)(w0 + m - 1 + kj) + dx;
        float y0f = floorf(py), x0f = floorf(px);
        int   y0  = (int)y0f,  x0i = (int)x0f;
        float wy1 = py - y0f,  wx1 = px - x0f;
        float wy0 = 1.0f - wy1, wx0 = 1.0f - wx1;
        int   y1  = y0 + 1,    x1i = x0i + 1;
        float vy0 = ((unsigned)y0  < (unsigned)H_) ? 1.f : 0.f;
        float vy1 = ((unsigned)y1  < (unsigned)H_) ? 1.f : 0.f;
        float vx0 = ((unsigned)x0i < (unsigned)W_) ? 1.f : 0.f;
        float vx1 = ((unsigned)x1i < (unsigned)W_) ? 1.f : 0.f;
        int yc0 = min(max(y0, 0), H_ - 1),  yc1 = min(max(y1, 0), H_ - 1);
        int xc0 = min(max(x0i, 0), W_ - 1), xc1 = min(max(x1i, 0), W_ - 1);
        float w00 = wy0 * wx0 * vy0 * vx0;
        float w01 = wy0 * wx1 * vy0 * vx1;
        float w10 = wy1 * wx0 * vy1 * vx0;
        float w11 = wy1 * wx1 * vy1 * vx1;
        int i00 = yc0 * W_ + xc0, i01 = yc0 * W_ + xc1;
        int i10 = yc1 * W_ + xc0, i11 = yc1 * W_ + xc1;
        const float* xb = x + (size_t)b * C_ * H_ * W_;
        for (int c = 0; c < C_; ++c) {
            const float* xc = xb + c * (H_ * W_);
            float v = w00 * xc[i00] + w01 * xc[i01] + w10 * xc[i10] + w11 * xc[i11];
            sA[(c * KK_ + kk) * TM_ + m] = v;
        }
    }
    __syncthreads();

    // ---- Phase B: per-wave 16x16 tile, K=576 via V_WMMA_F32_16X16X4_F32.
    //      Two independent accumulator chains over interleaved K-chunks to
    //      break the D->C accumulate dependency on the matrix pipe.
    const int lane = tid & 31;
    const int wave = tid >> 5;
    const int half = lane >> 4;   // lane group 0-15 vs 16-31
    const int mn   = lane & 15;   // M for A-frag, N for B/D-frag
    const int cout = wave * 16 + mn;
    const float* wrow = w_conv + (size_t)cout * KDIM_;  // B[k][n] = w_conv[n][k]

    v8f acc0 = {0.f, 0.f, 0.f, 0.f, 0.f, 0.f, 0.f, 0.f};
    v8f acc1 = {0.f, 0.f, 0.f, 0.f, 0.f, 0.f, 0.f, 0.f};
    for (int k = 0; k < KDIM_; k += 8) {
        int kb0 = k + 2 * half;        // chain 0: K-slice [k, k+4)
        int kb1 = k + 4 + 2 * half;    // chain 1: K-slice [k+4, k+8)
        v2f a0, a1, bb0, bb1;
        // A 16x4 f32 layout: lane m, VGPR0 = K(2*half), VGPR1 = K(2*half+1)
        a0.x = sA[(kb0 + 0) * TM_ + mn];
        a0.y = sA[(kb0 + 1) * TM_ + mn];
        a1.x = sA[(kb1 + 0) * TM_ + mn];
        a1.y = sA[(kb1 + 1) * TM_ + mn];
        // B 4x16 f32 layout: lane n, VGPR0 = row(2*half), VGPR1 = row(2*half+1)
        bb0 = *(const v2f*)(wrow + kb0);   // kb even -> 8B aligned
        bb1 = *(const v2f*)(wrow + kb1);
        acc0 = __builtin_amdgcn_wmma_f32_16x16x4_f32(
            false, a0, false, bb0, (short)0, acc0, false, false);
        acc1 = __builtin_amdgcn_wmma_f32_16x16x4_f32(
            false, a1, false, bb1, (short)0, acc1, false, false);
    }

    // ---- merge chains, bias + store:
    //      D lane layout -> lane n = cout, VGPR v -> pixel v + 8*half
    float bias = b_conv[cout];
    float* orow = out + (((size_t)b * CO_ + cout) * H_ + ho) * W_ + w0;
    #pragma unroll
    for (int v = 0; v < 8; ++v) {
        int m = v + 8 * half;
        orow[m] = (acc0[v] + acc1[v]) + bias;
    }
}

extern "C" void kernel_launch(void* const* d_in, const int* in_sizes, int n_in,
                              void* d_out, int out_size, void* d_ws, size_t ws_size,
                              hipStream_t stream) {
    const float* x      = (const float*)d_in[0];
    const float* w_off  = (const float*)d_in[1];
    const float* b_off  = (const float*)d_in[2];
    const float* w_conv = (const float*)d_in[3];
    const float* b_conv = (const float*)d_in[4];
    float* out = (float*)d_out;

    dim3 grid(B_ * H_ * (W_ / TM_));   // 32*64*4 = 8192 blocks
    deform_conv2d_wmma_f32<<<grid, 128, 0, stream>>>(x, w_off, b_off, w_conv, b_conv, out);
}